// LuongAttnDecoderRNN_52132313039325
// MI455X (gfx1250) — compile-verified
//
#include <hip/hip_runtime.h>
#include <hip/hip_bf16.h>
#include <math.h>

// ---------------------------------------------------------------------------
// LuongAttnDecoderRNN single step, fp32, MI455X (gfx1250).
// Memory-bound (~310MB traffic, ~13us floor @ 23.3TB/s). All GEMMs use
// V_WMMA_F32_16X16X4_F32 (native fp32 WMMA, wave32).
// Final V-GEMM: A staged in LDS (bank-conflict-free reads), W_out streamed
// from HBM exactly once; B prefetched one K-step ahead; all 8 A fragments
// issued before their WMMAs so LDS latency overlaps WMMA issue.
// ---------------------------------------------------------------------------

typedef float v2f __attribute__((ext_vector_type(2)));
typedef float v4f __attribute__((ext_vector_type(4)));
typedef float v8f __attribute__((ext_vector_type(8)));

#define BDIM 128
#define HDIM 1024
#define LDIM 128
#define VDIM 50000
#define INDIM 1088   // H + 32 + 32
#define G3H  3072    // 3*H

#define KC       256   // K-chunk staged in LDS for the V-GEMM
#define LDSTRIDE 260   // floats per LDS row; 260 % 64 == 4 -> conflict-free b64 reads

// -------------------------------------------------------------------
// Kernel 1: build GRU input x[b, 0:1088] = [emb[tok], sound_emb, num_emb]
// -------------------------------------------------------------------
__global__ void build_x_kernel(const int* __restrict__ seq,
                               const int* __restrict__ c2n,
                               const int* __restrict__ c2f,
                               const float* __restrict__ emb,
                               const float* __restrict__ semb,
                               const float* __restrict__ nemb,
                               float* __restrict__ x) {
    int b = blockIdx.x;
    int t = threadIdx.x;                    // 256 threads
    int tok = seq[b];
    const float* erow = emb + (size_t)tok * HDIM;
    float* xrow = x + (size_t)b * INDIM;
    for (int i = t; i < HDIM; i += 256) xrow[i] = erow[i];
    if (t < 32) {
        xrow[HDIM + t]      = semb[c2f[tok] * 32 + t];
        xrow[HDIM + 32 + t] = nemb[c2n[tok] * 32 + t];
    }
}

// -------------------------------------------------------------------
// Generic WMMA fp32 GEMM: one 16x16 D tile per wave, software-pipelined
// (next K-step's A/B fragments prefetched before this step's WMMA).
//   TRANSB=1: W stored [N,K] row-major (compute A @ W.T)
//   TRANSB=0: W stored [K,N] row-major (compute A @ W)
// M,N multiples of 16; K multiple of 4. 8 waves/block.
// -------------------------------------------------------------------
template <bool TRANSB, int ACT>
__global__ void wmma_gemm_tile(const float* __restrict__ A,
                               const float* __restrict__ W,
                               const float* __restrict__ bias,
                               float* __restrict__ D,
                               int M, int N, int K, int lda) {
    int gwave = (blockIdx.x * blockDim.x + threadIdx.x) >> 5;
    int lane  = threadIdx.x & 31;
    int ntn   = N >> 4;
    int ntile = (M >> 4) * ntn;
    if (gwave >= ntile) return;             // whole-wave uniform exit
    int tm = gwave / ntn;
    int tn = gwave - tm * ntn;

    int arow  = tm * 16 + (lane & 15);
    int bcol  = tn * 16 + (lane & 15);
    int khalf = (lane >> 4) * 2;

    const float* aptr = A + (size_t)arow * lda + khalf;
    v8f acc = {0.f, 0.f, 0.f, 0.f, 0.f, 0.f, 0.f, 0.f};

    if (TRANSB) {
        const float* wptr = W + (size_t)bcol * K + khalf;
        v2f ac = *(const v2f*)(aptr);
        v2f bc = *(const v2f*)(wptr);
        for (int k = 0; k < K; k += 4) {
            int kn = (k + 4 < K) ? (k + 4) : k;          // clamp: safe redundant load
            v2f an = *(const v2f*)(aptr + kn);
            v2f bn = *(const v2f*)(wptr + kn);
            acc = __builtin_amdgcn_wmma_f32_16x16x4_f32(false, ac, false, bc,
                                                        (short)0, acc, false, false);
            ac = an; bc = bn;
        }
    } else {
        v2f ac = *(const v2f*)(aptr);
        v2f bc;
        bc.x = W[(size_t)khalf * N + bcol];
        bc.y = W[(size_t)(khalf + 1) * N + bcol];
        for (int k = 0; k < K; k += 4) {
            int kn = (k + 4 < K) ? (k + 4) : k;
            v2f an = *(const v2f*)(aptr + kn);
            v2f bn;
            bn.x = W[(size_t)(kn + khalf) * N + bcol];
            bn.y = W[(size_t)(kn + khalf + 1) * N + bcol];
            acc = __builtin_amdgcn_wmma_f32_16x16x4_f32(false, ac, false, bc,
                                                        (short)0, acc, false, false);
            ac = an; bc = bn;
        }
    }

    float bv = bias ? bias[bcol] : 0.0f;
    int rbase = tm * 16 + (lane >> 4) * 8;
#pragma unroll
    for (int j = 0; j < 8; ++j) {
        float v = acc[j] + bv;
        if (ACT == 1) v = tanhf(v);
        D[(size_t)(rbase + j) * N + bcol] = v;
    }
}

// -------------------------------------------------------------------
// Kernel: GRU gate fusion (PyTorch order r,z,n).
// -------------------------------------------------------------------
__global__ void gru_fuse_kernel(const float* __restrict__ gi,
                                const float* __restrict__ gh,
                                const float* __restrict__ hprev,
                                float* __restrict__ hcat,
                                float* __restrict__ hid_out) {
    int idx = blockIdx.x * blockDim.x + threadIdx.x;   // 0 .. B*H-1
    int b = idx >> 10, j = idx & (HDIM - 1);
    const float* gib = gi + (size_t)b * G3H;
    const float* ghb = gh + (size_t)b * G3H;
    float r = 1.f / (1.f + __expf(-(gib[j] + ghb[j])));
    float z = 1.f / (1.f + __expf(-(gib[HDIM + j] + ghb[HDIM + j])));
    float n = tanhf(gib[2 * HDIM + j] + r * ghb[2 * HDIM + j]);
    float h = (1.f - z) * n + z * hprev[idx];
    hcat[(size_t)b * (2 * HDIM) + j] = h;
    hid_out[idx] = h;
}

// -------------------------------------------------------------------
// energies[b,l] = dot(q[b,:], enc[l,b,:])   (b_attn cancels in softmax)
// -------------------------------------------------------------------
__global__ void energies_kernel(const float* __restrict__ q,
                                const float* __restrict__ enc,
                                float* __restrict__ e) {
    int l = blockIdx.x, b = blockIdx.y;
    const float* er = enc + ((size_t)l * BDIM + b) * HDIM;
    const float* qr = q + (size_t)b * HDIM;
    float s = 0.f;
    for (int i = threadIdx.x; i < HDIM; i += 256) s += qr[i] * er[i];
    __shared__ float sm[256];
    sm[threadIdx.x] = s;
    __syncthreads();
    for (int off = 128; off > 0; off >>= 1) {
        if (threadIdx.x < off) sm[threadIdx.x] += sm[threadIdx.x + off];
        __syncthreads();
    }
    if (threadIdx.x == 0) e[(size_t)b * LDIM + l] = sm[0];
}

// -------------------------------------------------------------------
// Row softmax over L=128. In-place on ws energies + write d_out section.
// -------------------------------------------------------------------
__global__ void softmax_kernel(float* __restrict__ e,
                               float* __restrict__ attn_out) {
    int b = blockIdx.x, t = threadIdx.x;
    float v = e[(size_t)b * LDIM + t];
    __shared__ float sm[128];
    sm[t] = v;
    __syncthreads();
    for (int off = 64; off > 0; off >>= 1) {
        if (t < off) sm[t] = fmaxf(sm[t], sm[t + off]);
        __syncthreads();
    }
    float mx = sm[0];
    __syncthreads();
    float ex = __expf(v - mx);
    sm[t] = ex;
    __syncthreads();
    for (int off = 64; off > 0; off >>= 1) {
        if (t < off) sm[t] += sm[t + off];
        __syncthreads();
    }
    float w = ex / sm[0];
    e[(size_t)b * LDIM + t] = w;
    attn_out[(size_t)b * LDIM + t] = w;
}

// -------------------------------------------------------------------
// context[b,h] = sum_l w[b,l] * enc[l,b,h] -> hcat[:, H:2H]
// -------------------------------------------------------------------
__global__ void context_kernel(const float* __restrict__ w,
                               const float* __restrict__ enc,
                               float* __restrict__ hcat) {
    int b = blockIdx.y;
    int h = blockIdx.x * 256 + threadIdx.x;
    __shared__ float wl[LDIM];
    if (threadIdx.x < LDIM) wl[threadIdx.x] = w[(size_t)b * LDIM + threadIdx.x];
    __syncthreads();
    float s = 0.f;
    for (int l = 0; l < LDIM; ++l)
        s += wl[l] * enc[((size_t)l * BDIM + b) * HDIM + h];
    hcat[(size_t)b * (2 * HDIM) + HDIM + h] = s;
}

// -------------------------------------------------------------------
// Final GEMM: out[B,V] = co[B,1024] @ W_out[V,1024].T + b_out.
//  * Block = 8 waves, each wave one 16-column V-tile holding all 8 M-tiles
//    (64 acc VGPRs) -> W_out streamed from HBM exactly once.
//  * A staged per-block in LDS in K-chunks of 256 (130KB, row stride 260
//    floats => the 32 lanes' b64 A-reads cover all 64 banks, conflict-free).
//  * All 8 A fragments issued before the 8 WMMAs (batched ds loads ->
//    descending dscnt waits); B prefetched one K-step ahead.
// -------------------------------------------------------------------
__global__ void wmma_out_gemm(const float* __restrict__ A,     // [128,1024]
                              const float* __restrict__ W,     // [V,1024]
                              const float* __restrict__ bias,  // [V]
                              float* __restrict__ D) {
    extern __shared__ float As[];           // 128 * LDSTRIDE floats

    int wave = threadIdx.x >> 5;
    int lane = threadIdx.x & 31;
    int tn   = blockIdx.x * 8 + wave;
    bool active = (tn < (VDIM / 16));       // wave-uniform
    int l15   = lane & 15;
    int khalf = (lane >> 4) * 2;
    int col   = tn * 16 + l15;

    const float* wrow = active ? (W + (size_t)col * HDIM + khalf) : W;

    // staging mapping: 2 threads per row, 128 floats each, b128 chunks
    int srow = threadIdx.x >> 1;
    int scol = (threadIdx.x & 1) * 128;
    const float* gsrc0 = A + (size_t)srow * HDIM + scol;
    float*       ldst  = As + srow * LDSTRIDE + scol;

    // per-lane LDS read base (float index); row mt*16+l15, col k+khalf
    const float* abase = As + l15 * LDSTRIDE + khalf;

    v8f acc[8];
#pragma unroll
    for (int mt = 0; mt < 8; ++mt)
        acc[mt] = (v8f){0.f, 0.f, 0.f, 0.f, 0.f, 0.f, 0.f, 0.f};

    for (int kc = 0; kc < HDIM; kc += KC) {
        // ---- stage A chunk [128 x KC] into LDS (all 256 threads) ----
        const float* gsrc = gsrc0 + kc;
#pragma unroll
        for (int i = 0; i < 128; i += 4)
            *(v4f*)(ldst + i) = *(const v4f*)(gsrc + i);
        __syncthreads();

        if (active) {
            __builtin_prefetch(wrow + kc + KC, 0, 1);   // next chunk of W stream
            v2f bc = *(const v2f*)(wrow + kc);
            for (int k = 0; k < KC; k += 4) {
                int kn = (k + 4 < KC) ? (k + 4) : k;    // clamped prefetch
                v2f bn = *(const v2f*)(wrow + kc + kn);
                // issue all 8 A-fragment LDS loads before any WMMA
                v2f a[8];
#pragma unroll
                for (int mt = 0; mt < 8; ++mt)
                    a[mt] = *(const v2f*)(abase + mt * (16 * LDSTRIDE) + k);
#pragma unroll
                for (int mt = 0; mt < 8; ++mt)
                    acc[mt] = __builtin_amdgcn_wmma_f32_16x16x4_f32(
                        false, a[mt], false, bc, (short)0, acc[mt], false, false);
                bc = bn;
            }
        }
        __syncthreads();
    }

    if (active) {
        float bv = bias[col];
        int rsub = (lane >> 4) * 8;
#pragma unroll
        for (int mt = 0; mt < 8; ++mt)
#pragma unroll
            for (int j = 0; j < 8; ++j)
                D[(size_t)(mt * 16 + rsub + j) * VDIM + col] = acc[mt][j] + bv;
    }
}

// -------------------------------------------------------------------
extern "C" void kernel_launch(void* const* d_in, const int* in_sizes, int n_in,
                              void* d_out, int out_size, void* d_ws, size_t ws_size,
                              hipStream_t stream) {
    const int*   seq    = (const int*)d_in[0];
    const float* h0     = (const float*)d_in[1];    // [1,B,H]
    const float* enc    = (const float*)d_in[2];    // [L,B,H]
    const int*   c2n    = (const int*)d_in[3];
    const int*   c2f    = (const int*)d_in[4];
    const float* emb    = (const float*)d_in[5];
    const float* semb   = (const float*)d_in[6];
    const float* nemb   = (const float*)d_in[7];
    const float* W_ih   = (const float*)d_in[8];    // [3H, INDIM]
    const float* W_hh   = (const float*)d_in[9];    // [3H, H]
    const float* b_ih   = (const float*)d_in[10];
    const float* b_hh   = (const float*)d_in[11];
    const float* W_attn = (const float*)d_in[12];   // [H, H]
    const float* W_cat  = (const float*)d_in[14];   // [H, 2H]
    const float* b_cat  = (const float*)d_in[15];
    const float* W_out  = (const float*)d_in[16];   // [V, H]
    const float* b_out  = (const float*)d_in[17];

    float* out      = (float*)d_out;                          // [B,V]
    float* hid_out  = out + (size_t)BDIM * VDIM;              // [1,B,H]
    float* attn_out = hid_out + (size_t)BDIM * HDIM;          // [B,1,L]

    // workspace layout (floats)
    float* ws   = (float*)d_ws;
    float* x    = ws;                                         // [B, INDIM]
    float* gi   = x    + (size_t)BDIM * INDIM;                // [B, 3H]
    float* gh   = gi   + (size_t)BDIM * G3H;                  // [B, 3H]
    float* hcat = gh   + (size_t)BDIM * G3H;                  // [B, 2H]
    float* q    = hcat + (size_t)BDIM * 2 * HDIM;             // [B, H]
    float* eng  = q    + (size_t)BDIM * HDIM;                 // [B, L]
    float* co   = eng  + (size_t)BDIM * LDIM;                 // [B, H]

    // 1) gather embeddings -> x
    build_x_kernel<<<BDIM, 256, 0, stream>>>(seq, c2n, c2f, emb, semb, nemb, x);

    // 2) gi = x @ W_ih.T + b_ih
    wmma_gemm_tile<true, 0><<<(8 * (G3H / 16)) / 8, 256, 0, stream>>>(
        x, W_ih, b_ih, gi, BDIM, G3H, INDIM, INDIM);

    // 3) gh = h0 @ W_hh.T + b_hh
    wmma_gemm_tile<true, 0><<<(8 * (G3H / 16)) / 8, 256, 0, stream>>>(
        h0, W_hh, b_hh, gh, BDIM, G3H, HDIM, HDIM);

    // 4) GRU gates -> h_new (into hcat[:, :H] and d_out hidden)
    gru_fuse_kernel<<<(BDIM * HDIM) / 256, 256, 0, stream>>>(gi, gh, h0, hcat, hid_out);

    // 5) q = h_new @ W_attn
    wmma_gemm_tile<false, 0><<<(8 * (HDIM / 16)) / 8, 256, 0, stream>>>(
        hcat, W_attn, nullptr, q, BDIM, HDIM, HDIM, 2 * HDIM);

    // 6) energies
    energies_kernel<<<dim3(LDIM, BDIM), 256, 0, stream>>>(q, enc, eng);

    // 7) softmax -> attn_weights
    softmax_kernel<<<BDIM, LDIM, 0, stream>>>(eng, attn_out);

    // 8) context -> hcat[:, H:2H]
    context_kernel<<<dim3(HDIM / 256, BDIM), 256, 0, stream>>>(eng, enc, hcat);

    // 9) concat_out = tanh(hcat @ W_cat.T + b_cat)
    wmma_gemm_tile<true, 1><<<(8 * (HDIM / 16)) / 8, 256, 0, stream>>>(
        hcat, W_cat, b_cat, co, BDIM, HDIM, 2 * HDIM, 2 * HDIM);

    // 10) out = co @ W_out.T + b_out  (3125 V-tiles, 8 waves/block, LDS-staged A)
    int blocks = ((VDIM / 16) + 7) / 8;                 // 391
    size_t shbytes = (size_t)128 * LDSTRIDE * sizeof(float);  // 133,120 B
    wmma_out_gemm<<<blocks, 256, shbytes, stream>>>(co, W_out, b_out, out);
}